// TokenMixerLarge_1168231105051
// MI455X (gfx1250) — compile-verified
//
#include <hip/hip_runtime.h>
#include <math.h>

// ---------------------------------------------------------------------------
// Model constants (from the reference)
// ---------------------------------------------------------------------------
constexpr int B    = 4096;
constexpr int NS   = 26;
constexpr int ND   = 13;
constexpr int D    = 128;
constexpr int V    = 50000;
constexpr int H    = 8;
constexpr int E    = 8;           // router outputs; experts = E-1 = 7
constexpr int TOK  = B * 3;       // 12288 tokens (3 per sample)
constexpr int MIXK = 48;          // T * (D/H)
constexpr int MIXH = 192;
constexpr int MOEH = 512;
constexpr int DH   = 512;
constexpr float ALPHA = 2.0f;

typedef __attribute__((ext_vector_type(2))) float v2f;
typedef __attribute__((ext_vector_type(8))) float v8f;

// ---------------------------------------------------------------------------
// CDNA5 fp32 WMMA core: D(16x16,f32) = A(16x4,f32) x B(4x16,f32) + C
//   A layout: lanes 0-15 -> M=lane, VGPR0/1 = K {0,1}; lanes 16-31 -> K {2,3}
//   B layout: lanes 0-15 -> N=lane, VGPR0/1 = K {0,1}; lanes 16-31 -> K {2,3}
//   C layout: VGPR j -> M = j + 8*(lane>=16), N = lane&15
// ---------------------------------------------------------------------------
__device__ __forceinline__ v8f wmma4(v2f a, v2f b, v8f c) {
  return __builtin_amdgcn_wmma_f32_16x16x4_f32(false, a, false, b, (short)0, c,
                                               false, false);
}

// SiLU via hardware v_rcp_f32 (epilogue math; avoids IEEE div expansion)
__device__ __forceinline__ float siluf(float v) {
  return v * __builtin_amdgcn_rcpf(1.0f + __expf(-v));
}

// ---------------------------------------------------------------------------
// Generic N-strip GEMM: C[M, N] = act(A[M,K] @ W[K,N]); ACT 0=none 1=relu.
// One wave computes a 16 x (16*NT) strip, reusing the A fragment NT times
// (A traffic amortized 1/NT -> higher flop/byte on the WMMA pipe).
// 4 waves per 128-thread block (wave32).
// ---------------------------------------------------------------------------
template <int ACT, int NT>
__global__ void gemm_k(const float* __restrict__ Amat,
                       const float* __restrict__ Wmat, float* __restrict__ C,
                       int M, int K, int N) {
  const int lane = threadIdx.x & 31;
  const int wave = threadIdx.x >> 5;
  const int m0 = (blockIdx.y * 4 + wave) * 16;
  const int n0 = blockIdx.x * (16 * NT);
  if (m0 >= M) return;  // uniform per wave -> EXEC stays all-ones for WMMA
  const int ln = lane & 15;            // M index for A loads, N index for B/C
  const int ko = (lane >> 4) << 1;     // K sub-offset: 0 or 2
  const float* Ab = Amat + (size_t)m0 * K + (size_t)ln * K + ko;
  const float* Wb = Wmat + n0 + ln;
  v8f acc[NT];
  #pragma unroll
  for (int t = 0; t < NT; ++t) acc[t] = (v8f){};
  #pragma unroll 2
  for (int k = 0; k < K; k += 4) {
    const v2f a = *(const v2f*)(Ab + k);  // contiguous pair: one b64 load
    v2f b[NT];
    #pragma unroll
    for (int t = 0; t < NT; ++t) {
      b[t].x = Wb[(size_t)(k + ko) * N + 16 * t];
      b[t].y = Wb[(size_t)(k + ko + 1) * N + 16 * t];
    }
    #pragma unroll
    for (int t = 0; t < NT; ++t) acc[t] = wmma4(a, b[t], acc[t]);
  }
  float* Cb = C + (size_t)m0 * N + n0 + ln;
  const int mb = (lane >> 4) * 8;
  #pragma unroll
  for (int t = 0; t < NT; ++t) {
    #pragma unroll
    for (int j = 0; j < 8; ++j) {
      float v = acc[t][j];
      if (ACT == 1) v = fmaxf(v, 0.0f);
      Cb[(size_t)(mb + j) * N + 16 * t] = v;
    }
  }
}

// ---------------------------------------------------------------------------
// Fused SwiGLU up-projection: Hout[M,N] = (A@U) * silu(A@G)
// Dual accumulators per column tile, single pass over A.
// ---------------------------------------------------------------------------
template <int NT>
__global__ void gemm_swiglu_k(const float* __restrict__ Amat,
                              const float* __restrict__ U,
                              const float* __restrict__ G,
                              float* __restrict__ Hout, int M, int K, int N) {
  const int lane = threadIdx.x & 31;
  const int wave = threadIdx.x >> 5;
  const int m0 = (blockIdx.y * 4 + wave) * 16;
  const int n0 = blockIdx.x * (16 * NT);
  if (m0 >= M) return;
  const int ln = lane & 15;
  const int ko = (lane >> 4) << 1;
  const float* Ab = Amat + (size_t)m0 * K + (size_t)ln * K + ko;
  const float* Ub = U + n0 + ln;
  const float* Gb = G + n0 + ln;
  v8f au[NT], ag[NT];
  #pragma unroll
  for (int t = 0; t < NT; ++t) { au[t] = (v8f){}; ag[t] = (v8f){}; }
  for (int k = 0; k < K; k += 4) {
    const v2f a = *(const v2f*)(Ab + k);
    v2f bu[NT], bg[NT];
    #pragma unroll
    for (int t = 0; t < NT; ++t) {
      bu[t].x = Ub[(size_t)(k + ko) * N + 16 * t];
      bu[t].y = Ub[(size_t)(k + ko + 1) * N + 16 * t];
      bg[t].x = Gb[(size_t)(k + ko) * N + 16 * t];
      bg[t].y = Gb[(size_t)(k + ko + 1) * N + 16 * t];
    }
    #pragma unroll
    for (int t = 0; t < NT; ++t) {
      au[t] = wmma4(a, bu[t], au[t]);
      ag[t] = wmma4(a, bg[t], ag[t]);
    }
  }
  float* Hb = Hout + (size_t)m0 * N + n0 + ln;
  const int mb = (lane >> 4) * 8;
  #pragma unroll
  for (int t = 0; t < NT; ++t)
    #pragma unroll
    for (int j = 0; j < 8; ++j)
      Hb[(size_t)(mb + j) * N + 16 * t] = au[t][j] * siluf(ag[t][j]);
}

// ---------------------------------------------------------------------------
// MoE routed-expert GEMMs with row gather/scatter through per-expert buckets.
// Bucket count read is made wave-uniform (readfirstlane) so the early exit
// keeps EXEC all-ones around the WMMAs. NT=4 column strip (N=512 / N=128).
// ---------------------------------------------------------------------------
__global__ void moe_up_k(const float* __restrict__ Xn,
                         const float* __restrict__ U,
                         const float* __restrict__ G,
                         const int* __restrict__ list,
                         const int* __restrict__ cnt, int e,
                         float* __restrict__ Hout, int K, int N) {
  constexpr int NT = 4;
  const int lane = threadIdx.x & 31;
  const int wave = threadIdx.x >> 5;
  const int c = __builtin_amdgcn_readfirstlane(cnt[e]);
  const int m0 = (blockIdx.y * 4 + wave) * 16;
  const int n0 = blockIdx.x * (16 * NT);
  if (m0 >= c) return;
  const int ln = lane & 15;
  const int ko = (lane >> 4) << 1;
  const int r = m0 + ln;
  const int tokid = list[e * TOK + (r < c ? r : c - 1)];
  const float* Ab = Xn + (size_t)tokid * K + ko;
  const float* Ub = U + n0 + ln;
  const float* Gb = G + n0 + ln;
  v8f au[NT], ag[NT];
  #pragma unroll
  for (int t = 0; t < NT; ++t) { au[t] = (v8f){}; ag[t] = (v8f){}; }
  for (int k = 0; k < K; k += 4) {
    const v2f a = *(const v2f*)(Ab + k);
    v2f bu[NT], bg[NT];
    #pragma unroll
    for (int t = 0; t < NT; ++t) {
      bu[t].x = Ub[(size_t)(k + ko) * N + 16 * t];
      bu[t].y = Ub[(size_t)(k + ko + 1) * N + 16 * t];
      bg[t].x = Gb[(size_t)(k + ko) * N + 16 * t];
      bg[t].y = Gb[(size_t)(k + ko + 1) * N + 16 * t];
    }
    #pragma unroll
    for (int t = 0; t < NT; ++t) {
      au[t] = wmma4(a, bu[t], au[t]);
      ag[t] = wmma4(a, bg[t], ag[t]);
    }
  }
  const int mb = (lane >> 4) * 8;
  #pragma unroll
  for (int j = 0; j < 8; ++j) {
    const int rr = m0 + mb + j;
    if (rr < c) {
      float* __restrict__ Ho = Hout + (size_t)rr * N + n0 + ln;
      #pragma unroll
      for (int t = 0; t < NT; ++t) Ho[16 * t] = au[t][j] * siluf(ag[t][j]);
    }
  }
}

__global__ void moe_down_k(const float* __restrict__ Hid,
                           const float* __restrict__ Dw,
                           const int* __restrict__ list,
                           const int* __restrict__ cnt,
                           const float* __restrict__ coef, int e,
                           float* __restrict__ Out, int K, int N) {
  constexpr int NT = 4;
  const int lane = threadIdx.x & 31;
  const int wave = threadIdx.x >> 5;
  const int c = __builtin_amdgcn_readfirstlane(cnt[e]);
  const int m0 = (blockIdx.y * 4 + wave) * 16;
  const int n0 = blockIdx.x * (16 * NT);
  if (m0 >= c) return;
  const int ln = lane & 15;
  const int ko = (lane >> 4) << 1;
  const int r = m0 + ln;
  const int ar = r < c ? r : c - 1;  // clamp: rows past count are discarded
  const float* Ab = Hid + (size_t)ar * K + ko;
  const float* Wb = Dw + n0 + ln;
  v8f acc[NT];
  #pragma unroll
  for (int t = 0; t < NT; ++t) acc[t] = (v8f){};
  #pragma unroll 2
  for (int k = 0; k < K; k += 4) {
    const v2f a = *(const v2f*)(Ab + k);
    v2f b[NT];
    #pragma unroll
    for (int t = 0; t < NT; ++t) {
      b[t].x = Wb[(size_t)(k + ko) * N + 16 * t];
      b[t].y = Wb[(size_t)(k + ko + 1) * N + 16 * t];
    }
    #pragma unroll
    for (int t = 0; t < NT; ++t) acc[t] = wmma4(a, b[t], acc[t]);
  }
  const int mb = (lane >> 4) * 8;
  #pragma unroll
  for (int j = 0; j < 8; ++j) {
    const int rr = m0 + mb + j;
    if (rr < c) {
      const int tokid = list[e * TOK + rr];
      const float cf = coef[tokid];
      float* dst = Out + (size_t)tokid * N + n0 + ln;
      #pragma unroll
      for (int t = 0; t < NT; ++t)
        dst[16 * t] += cf * acc[t][j];  // tokens unique per expert: race-free
    }
  }
}

// ---------------------------------------------------------------------------
// Elementwise / reduction helpers (wave32)
// ---------------------------------------------------------------------------
__global__ void k_gather_sp(const int* __restrict__ sp,
                            const float* __restrict__ emb,
                            float* __restrict__ Xsp) {
  size_t i = (size_t)blockIdx.x * blockDim.x + threadIdx.x;
  if (i >= (size_t)B * NS * D) return;
  int d = (int)(i % D);
  int s = (int)((i / D) % NS);
  int b = (int)(i / ((size_t)D * NS));
  int id = sp[b * NS + s];
  Xsp[i] = emb[((size_t)s * V + id) * D + d];
}

__global__ void k_build_dn(const float* __restrict__ dn,
                           const float* __restrict__ wdense,
                           float* __restrict__ Xdn) {
  size_t i = (size_t)blockIdx.x * blockDim.x + threadIdx.x;
  if (i >= (size_t)B * ND * D) return;
  int d = (int)(i % D);
  int j = (int)((i / D) % ND);
  int b = (int)(i / ((size_t)D * ND));
  Xdn[i] = dn[b * ND + j] * wdense[j * D + d];
}

__global__ void k_concat2(const float* __restrict__ ts,
                          const float* __restrict__ td,
                          float* __restrict__ Xg) {
  int i = blockIdx.x * blockDim.x + threadIdx.x;
  if (i >= B * 2 * D) return;
  int d = i % (2 * D);
  int b = i / (2 * D);
  Xg[i] = (d < D) ? ts[b * D + d] : td[b * D + d - D];
}

__global__ void k_build_x(const float* __restrict__ g,
                          const float* __restrict__ ts,
                          const float* __restrict__ td, float* __restrict__ x) {
  int i = blockIdx.x * blockDim.x + threadIdx.x;
  if (i >= TOK * D) return;
  int r = i % (3 * D);
  int b = i / (3 * D);
  float v;
  if (r < D) v = g[b * D + r];
  else if (r < 2 * D) v = ts[b * D + r - D];
  else v = td[b * D + r - 2 * D];
  x[i] = v;
}

__global__ void k_rms(const float* __restrict__ X, const float* __restrict__ sc,
                      float* __restrict__ Y, int rows) {
  const int lane = threadIdx.x & 31;
  const int row = blockIdx.x * (blockDim.x >> 5) + (threadIdx.x >> 5);
  if (row >= rows) return;
  const float4 v = ((const float4*)(X + (size_t)row * D))[lane];
  float ss = v.x * v.x + v.y * v.y + v.z * v.z + v.w * v.w;
  #pragma unroll
  for (int off = 16; off; off >>= 1) ss += __shfl_xor(ss, off, 32);
  const float r = rsqrtf(ss * (1.0f / D) + 1e-6f);
  const float4 s4 = ((const float4*)sc)[lane];
  float4 o;
  o.x = v.x * s4.x * r; o.y = v.y * s4.y * r;
  o.z = v.z * s4.z * r; o.w = v.w * s4.w * r;
  ((float4*)(Y + (size_t)row * D))[lane] = o;
}

// x[row] += rms(y[row], sc)
__global__ void k_rms_add(float* __restrict__ Xio, const float* __restrict__ Yv,
                          const float* __restrict__ sc, int rows) {
  const int lane = threadIdx.x & 31;
  const int row = blockIdx.x * (blockDim.x >> 5) + (threadIdx.x >> 5);
  if (row >= rows) return;
  const float4 v = ((const float4*)(Yv + (size_t)row * D))[lane];
  float ss = v.x * v.x + v.y * v.y + v.z * v.z + v.w * v.w;
  #pragma unroll
  for (int off = 16; off; off >>= 1) ss += __shfl_xor(ss, off, 32);
  const float r = rsqrtf(ss * (1.0f / D) + 1e-6f);
  const float4 s4 = ((const float4*)sc)[lane];
  float4 o = ((float4*)(Xio + (size_t)row * D))[lane];
  o.x += v.x * s4.x * r; o.y += v.y * s4.y * r;
  o.z += v.z * s4.z * r; o.w += v.w * s4.w * r;
  ((float4*)(Xio + (size_t)row * D))[lane] = o;
}

// (B,3,H,16) -> (H, B, 3*16): xs[h*B+b, t*16+c] = xn[b, t, h*16+c]
__global__ void k_perm_xs(const float* __restrict__ xn, float* __restrict__ xs) {
  int i = blockIdx.x * blockDim.x + threadIdx.x;
  if (i >= TOK * D) return;
  int j = i % MIXK;
  int b = (i / MIXK) % B;
  int h = i / (MIXK * B);
  xs[i] = xn[(size_t)b * (3 * D) + (j / 16) * D + h * 16 + (j % 16)];
}

// inverse: xr[b, t, h*16+c] = xm[h*B+b, t*16+c]
__global__ void k_perm_xr(const float* __restrict__ xm, float* __restrict__ xr) {
  int i = blockIdx.x * blockDim.x + threadIdx.x;
  if (i >= TOK * D) return;
  int dd = i % D;
  int t = (i / D) % 3;
  int b = i / (3 * D);
  int h = dd / 16, c = dd % 16;
  xr[i] = xm[(size_t)(h * B + b) * MIXK + t * 16 + c];
}

__global__ void k_router(const float* __restrict__ Xn,
                         const float* __restrict__ Wr, int* __restrict__ expertI,
                         float* __restrict__ coef) {
  const int lane = threadIdx.x & 31;
  const int tok = blockIdx.x * (blockDim.x >> 5) + (threadIdx.x >> 5);
  if (tok >= TOK) return;
  float l = -3.0e38f;
  if (lane < E) {
    float s = 0.f;
    const float* xr = Xn + (size_t)tok * D;
    for (int j = 0; j < D; ++j) s += xr[j] * Wr[j * E + lane];
    l = s;
  }
  float m = l;
  #pragma unroll
  for (int off = 16; off; off >>= 1) m = fmaxf(m, __shfl_xor(m, off, 32));
  float ex = (lane < E) ? __expf(l - m) : 0.0f;
  float s = ex;
  #pragma unroll
  for (int off = 16; off; off >>= 1) s += __shfl_xor(s, off, 32);
  unsigned long long bal = __ballot(l == m);  // lowest lane wins ties (jax top_k)
  int idx = __ffsll((unsigned long long)bal) - 1;
  if (lane == 0) {
    expertI[tok] = idx;           // top-1 prob = 1/s after softmax shift
    coef[tok] = ALPHA * __builtin_amdgcn_rcpf(s);
  }
}

__global__ void k_zero_cnt(int* __restrict__ cnt) {
  if (threadIdx.x < E) cnt[threadIdx.x] = 0;
}

__global__ void k_bucket(const int* __restrict__ expertI, int* __restrict__ cnt,
                         int* __restrict__ list) {
  int t = blockIdx.x * blockDim.x + threadIdx.x;
  if (t >= TOK) return;
  int e = expertI[t];
  if (e >= 0 && e < E - 1) {
    int p = atomicAdd(&cnt[e], 1);
    list[e * TOK + p] = t;
  }
}

__global__ void k_add(float* __restrict__ x, const float* __restrict__ y, int n) {
  int i = blockIdx.x * blockDim.x + threadIdx.x;
  if (i < n) x[i] += y[i];
}

__global__ void k_copy(float* __restrict__ dst, const float* __restrict__ src,
                       int n) {
  int i = blockIdx.x * blockDim.x + threadIdx.x;
  if (i < n) dst[i] = src[i];
}

__global__ void k_mean(const float* __restrict__ x, float* __restrict__ h) {
  int i = blockIdx.x * blockDim.x + threadIdx.x;
  if (i >= B * D) return;
  int d = i % D, b = i / D;
  const float* xb = x + (size_t)b * 3 * D;
  h[i] = (xb[d] + xb[D + d] + xb[2 * D + d]) * (1.0f / 3.0f);
}

__global__ void k_headout(const float* __restrict__ H2,
                          const float* __restrict__ Wo, float* __restrict__ Out) {
  const int lane = threadIdx.x & 31;
  const int row = blockIdx.x * (blockDim.x >> 5) + (threadIdx.x >> 5);
  if (row >= B) return;
  float s = 0.f;
  for (int j = lane; j < DH; j += 32) s += H2[(size_t)row * DH + j] * Wo[j];
  #pragma unroll
  for (int off = 16; off; off >>= 1) s += __shfl_xor(s, off, 32);
  if (lane == 0) Out[row] = s;
}

// ---------------------------------------------------------------------------
// Host-side orchestration
// ---------------------------------------------------------------------------
static inline dim3 gemm_grid(int M, int N, int NT) {
  return dim3(N / (16 * NT), (M + 63) / 64);
}

extern "C" void kernel_launch(void* const* d_in, const int* in_sizes, int n_in,
                              void* d_out, int out_size, void* d_ws,
                              size_t ws_size, hipStream_t stream) {
  (void)in_sizes; (void)n_in; (void)out_size; (void)ws_size;

  // ---- input pointers (insertion-order DFS flatten of setup_inputs()) ----
  const int*   sp     = (const int*)d_in[0];
  const float* dn     = (const float*)d_in[1];
  const float* emb    = (const float*)d_in[2];
  const float* wdense = (const float*)d_in[3];
  const float* ws1 = (const float*)d_in[4];
  const float* ws2 = (const float*)d_in[5];
  const float* wd1 = (const float*)d_in[6];
  const float* wd2 = (const float*)d_in[7];
  const float* wg1 = (const float*)d_in[8];
  const float* wg2 = (const float*)d_in[9];
  // blocks: 16 leaves each, starting at 10
  const float* hw0 = (const float*)d_in[74];
  const float* hw1 = (const float*)d_in[75];
  const float* hwo = (const float*)d_in[76];

  // ---- workspace layout (fp32 elements; int buffers share 4-byte slots) ----
  float* Wb = (float*)d_ws;
  size_t o = 0;
  auto alloc = [&](size_t n) { float* p = Wb + o; o += n; return p; };
  float* R0 = alloc((size_t)B * NS * D);  // Xsp; later: hid + hidR
  float* R1 = alloc((size_t)B * ND * D);  // Xdn; later: hidM
  float* t1   = alloc((size_t)B * D);
  float* ts   = alloc((size_t)B * D);
  float* td   = alloc((size_t)B * D);
  float* gtok = alloc((size_t)B * D);
  float* Xg   = alloc((size_t)B * 2 * D);
  float* x    = alloc((size_t)TOK * D);
  float* xn   = alloc((size_t)TOK * D);
  float* xs   = alloc((size_t)TOK * D);
  float* xm   = alloc((size_t)TOK * D);
  float* xr   = alloc((size_t)TOK * D);
  float* ybuf = alloc((size_t)TOK * D);
  float* outb = alloc((size_t)TOK * D);
  float* cach = alloc((size_t)TOK * D);
  float* hbuf = alloc((size_t)B * D);
  float* h1   = alloc((size_t)B * DH);
  float* h2   = alloc((size_t)B * DH);
  float* coef = alloc((size_t)TOK);
  int* expertI = (int*)alloc((size_t)TOK);
  int* cnt     = (int*)alloc(8);
  int* list    = (int*)alloc((size_t)(E - 1) * TOK);
  float* Xsp = R0;
  float* Xdn = R1;
  float* hid  = R0;                       // TOK*MOEH
  float* hidR = R0 + (size_t)TOK * MOEH;  // TOK*MOEH (fits in Xsp region)
  float* hidM = R1;                       // (H*B)*MIXH (fits in Xdn region)

  const dim3 blk128(128), blk256(256);

  // ---- tokenization ----
  k_gather_sp<<<(B * NS * D + 255) / 256, blk256, 0, stream>>>(sp, emb, Xsp);
  k_build_dn<<<(B * ND * D + 255) / 256, blk256, 0, stream>>>(dn, wdense, Xdn);
  gemm_k<1, 4><<<gemm_grid(B, D, 4), blk128, 0, stream>>>(Xsp, ws1, t1, B,
                                                          NS * D, D);
  gemm_k<0, 4><<<gemm_grid(B, D, 4), blk128, 0, stream>>>(t1, ws2, ts, B, D, D);
  gemm_k<1, 4><<<gemm_grid(B, D, 4), blk128, 0, stream>>>(Xdn, wd1, t1, B,
                                                          ND * D, D);
  gemm_k<0, 4><<<gemm_grid(B, D, 4), blk128, 0, stream>>>(t1, wd2, td, B, D, D);
  k_concat2<<<(B * 2 * D + 255) / 256, blk256, 0, stream>>>(ts, td, Xg);
  gemm_k<1, 4><<<gemm_grid(B, D, 4), blk128, 0, stream>>>(Xg, wg1, t1, B, 2 * D,
                                                          D);
  gemm_k<0, 4><<<gemm_grid(B, D, 4), blk128, 0, stream>>>(t1, wg2, gtok, B, D,
                                                          D);
  k_build_x<<<(TOK * D + 255) / 256, blk256, 0, stream>>>(gtok, ts, td, x);

  // ---- blocks ----
  const int rowBlocksTok = (TOK + 7) / 8;  // k_rms & router: 8 rows / 256-thr blk
  for (int l = 0; l < 4; ++l) {
    const int IB = 10 + 16 * l;
    const float* n1  = (const float*)d_in[IB + 0];
    const float* mu  = (const float*)d_in[IB + 1];
    const float* mg  = (const float*)d_in[IB + 2];
    const float* md  = (const float*)d_in[IB + 3];
    const float* n2  = (const float*)d_in[IB + 4];
    const float* ru  = (const float*)d_in[IB + 5];
    const float* rg  = (const float*)d_in[IB + 6];
    const float* rd  = (const float*)d_in[IB + 7];
    const float* nm  = (const float*)d_in[IB + 8];
    const float* wr  = (const float*)d_in[IB + 9];
    const float* eu  = (const float*)d_in[IB + 10];
    const float* eg  = (const float*)d_in[IB + 11];
    const float* ed  = (const float*)d_in[IB + 12];
    const float* su  = (const float*)d_in[IB + 13];
    const float* sg  = (const float*)d_in[IB + 14];
    const float* sd  = (const float*)d_in[IB + 15];

    // MixingReverting
    k_rms<<<rowBlocksTok, blk256, 0, stream>>>(x, n1, xn, TOK);
    k_perm_xs<<<(TOK * D + 255) / 256, blk256, 0, stream>>>(xn, xs);
    gemm_swiglu_k<4><<<gemm_grid(H * B, MIXH, 4), blk128, 0, stream>>>(
        xs, mu, mg, hidM, H * B, MIXK, MIXH);
    gemm_k<0, 3><<<gemm_grid(H * B, MIXK, 3), blk128, 0, stream>>>(
        hidM, md, xm, H * B, MIXH, MIXK);
    k_perm_xr<<<(TOK * D + 255) / 256, blk256, 0, stream>>>(xm, xr);
    gemm_swiglu_k<4><<<gemm_grid(TOK, MOEH, 4), blk128, 0, stream>>>(
        xr, ru, rg, hid, TOK, D, MOEH);
    gemm_k<0, 4><<<gemm_grid(TOK, D, 4), blk128, 0, stream>>>(hid, rd, ybuf,
                                                              TOK, MOEH, D);
    k_rms_add<<<rowBlocksTok, blk256, 0, stream>>>(x, ybuf, n2, TOK);

    // SparsePertokenMoE (top-1 routing; reference loops range(TOPK-1)==1)
    k_rms<<<rowBlocksTok, blk256, 0, stream>>>(x, nm, xn, TOK);
    k_router<<<rowBlocksTok, blk256, 0, stream>>>(xn, wr, expertI, coef);
    k_zero_cnt<<<1, 32, 0, stream>>>(cnt);
    k_bucket<<<(TOK + 255) / 256, blk256, 0, stream>>>(expertI, cnt, list);
    // shared expert -> writes all of outb
    gemm_swiglu_k<4><<<gemm_grid(TOK, MOEH, 4), blk128, 0, stream>>>(
        xn, su, sg, hid, TOK, D, MOEH);
    gemm_k<0, 4><<<gemm_grid(TOK, D, 4), blk128, 0, stream>>>(hid, sd, outb,
                                                              TOK, MOEH, D);
    // routed experts: gathered GEMMs, worst-case grid, count-based early exit
    for (int e = 0; e < E - 1; ++e) {
      moe_up_k<<<gemm_grid(TOK, MOEH, 4), blk128, 0, stream>>>(
          xn, eu + (size_t)e * D * MOEH, eg + (size_t)e * D * MOEH, list, cnt,
          e, hidR, D, MOEH);
      moe_down_k<<<gemm_grid(TOK, D, 4), blk128, 0, stream>>>(
          hidR, ed + (size_t)e * MOEH * D, list, cnt, coef, e, outb, MOEH, D);
    }
    k_add<<<(TOK * D + 255) / 256, blk256, 0, stream>>>(x, outb, TOK * D);

    // inter-block skip: save after blocks 0,2; add after blocks 1,3
    if (l == 1 || l == 3)
      k_add<<<(TOK * D + 255) / 256, blk256, 0, stream>>>(x, cach, TOK * D);
    if (l == 0 || l == 2)
      k_copy<<<(TOK * D + 255) / 256, blk256, 0, stream>>>(cach, x, TOK * D);
  }

  // ---- head ----
  k_mean<<<(B * D + 255) / 256, blk256, 0, stream>>>(x, hbuf);
  gemm_k<1, 4><<<gemm_grid(B, DH, 4), blk128, 0, stream>>>(hbuf, hw0, h1, B, D,
                                                           DH);
  gemm_k<1, 4><<<gemm_grid(B, DH, 4), blk128, 0, stream>>>(h1, hw1, h2, B, DH,
                                                           DH);
  k_headout<<<(B + 7) / 8, blk256, 0, stream>>>(h2, hwo, (float*)d_out);
}